// HMMFoldingModel_39565238731448
// MI455X (gfx1250) — compile-verified
//
#include <hip/hip_runtime.h>
#include <hip/hip_bf16.h>

typedef __attribute__((ext_vector_type(2))) float v2f;
typedef __attribute__((ext_vector_type(8))) float v8f;

#if defined(__AMDGCN__) && __has_builtin(__builtin_amdgcn_wmma_f32_16x16x4_f32)
#define HMM_USE_WMMA 1
#else
#define HMM_USE_WMMA 0
#endif

namespace {
constexpr int Bn = 256;    // batch
constexpr int Tn = 4096;   // time
constexpr int An = 26;     // alphabet
constexpr int Ln = 64;     // chunk length
constexpr int Cn = Tn / Ln;
constexpr float NEGV = -1e9f;

// workspace layout (float offsets)
constexpr long long PAR_BMAT  = 0;     // 4*26 emission distributions
constexpr long long PAR_LOGA  = 112;   // 16   masked log-transition
constexpr long long PAR_LOGPI = 128;   // 4
constexpr long long PAR_WOUT  = 132;   // 20
constexpr long long OFF_LE    = 256;                          // log_emit  [B*T*4]
constexpr long long OFF_AL    = OFF_LE + (long long)Bn*Tn*4;  // alphas    [B*T*4]
constexpr long long OFF_R     = OFF_AL + (long long)Bn*Tn*4;  // chunk mats[B*C*16]
constexpr long long OFF_SA    = OFF_R  + (long long)Bn*Cn*16; // startAlpha[B*C*4]
constexpr long long OFF_SB    = OFF_SA + (long long)Bn*Cn*4;  // startBeta [B*C*4]
}

__device__ __forceinline__ float lse4(float a, float b, float c, float d) {
  float m = fmaxf(fmaxf(a, b), fmaxf(c, d));
  return m + __logf(__expf(a - m) + __expf(b - m) + __expf(c - m) + __expf(d - m));
}

// ---------------------------------------------------------------- parameters
__global__ void k_setup(const float* __restrict__ emit, const float* __restrict__ trans,
                        const float* __restrict__ init, const float* __restrict__ wout,
                        float* __restrict__ par) {
  if (threadIdx.x != 0 || blockIdx.x != 0) return;
  for (int s = 0; s < 4; ++s) {                    // Bmat = softmax(emit_logits)
    float m = emit[s * An];
    for (int a = 1; a < An; ++a) m = fmaxf(m, emit[s * An + a]);
    float sum = 0.f;
    for (int a = 0; a < An; ++a) sum += expf(emit[s * An + a] - m);
    float ls = m + logf(sum);
    for (int a = 0; a < An; ++a) par[PAR_BMAT + s * An + a] = expf(emit[s * An + a] - ls);
  }
  const bool mask[16] = {1,1,1,0, 0,1,1,1, 0,0,1,1, 0,0,0,1};
  for (int i = 0; i < 4; ++i) {                    // masked log_softmax rows
    float v[4];
    for (int j = 0; j < 4; ++j) v[j] = mask[i*4+j] ? trans[i*4+j] : NEGV;
    float m  = fmaxf(fmaxf(v[0], v[1]), fmaxf(v[2], v[3]));
    float ls = m + logf(expf(v[0]-m)+expf(v[1]-m)+expf(v[2]-m)+expf(v[3]-m));
    for (int j = 0; j < 4; ++j) par[PAR_LOGA + i*4 + j] = mask[i*4+j] ? (v[j]-ls) : NEGV;
  }
  {
    float m  = fmaxf(fmaxf(init[0], init[1]), fmaxf(init[2], init[3]));
    float ls = m + logf(expf(init[0]-m)+expf(init[1]-m)+expf(init[2]-m)+expf(init[3]-m));
    for (int s = 0; s < 4; ++s) par[PAR_LOGPI + s] = init[s] - ls;
  }
  for (int i = 0; i < 20; ++i) par[PAR_WOUT + i] = wout[i];
}

// ------------------------------------------------- emission log-likelihoods
// 256 positions / block. Async global->LDS staging of x (ASYNCcnt path),
// per-lane softmax numerators, then the [16x26]x[26x4] dot via 7 accumulating
// V_WMMA_F32_16X16X4_F32 per 16-position tile.
__global__ __launch_bounds__(256) void k_emission(const float* __restrict__ x,
                                                  const float* __restrict__ par,
                                                  float* __restrict__ le) {
  __shared__ float sP[256 * An];   // x tile, then exp numerators in place
  __shared__ float sBT[28 * 16];   // B fragment source: BT[k][n] = Bmat[n][k]
  __shared__ float sInv[256];      // 1 / sum_exp per position
  const int tid = threadIdx.x;
  const long long posBase = (long long)blockIdx.x * 256;
  const float4* src = reinterpret_cast<const float4*>(x + posBase * An);

#if defined(__AMDGCN__)
  // stage 256*26 floats = 1664 float4 via async copies; LDS dest offset is the
  // low 32 bits of the flat shared-aperture address.
  {
    const unsigned lbase = (unsigned)(uintptr_t)(void*)sP;
#pragma unroll
    for (int i = 0; i < 6; ++i) {                 // 6*256 = 1536
      const int idx = tid + i * 256;
      const unsigned loff = lbase + (unsigned)idx * 16u;
      const unsigned long long g = (unsigned long long)(uintptr_t)(src + idx);
      asm volatile("global_load_async_to_lds_b128 %0, %1, off"
                   :: "v"(loff), "v"(g) : "memory");
    }
    if (tid < 128) {                              // remaining 128
      const int idx = tid + 1536;
      const unsigned loff = lbase + (unsigned)idx * 16u;
      const unsigned long long g = (unsigned long long)(uintptr_t)(src + idx);
      asm volatile("global_load_async_to_lds_b128 %0, %1, off"
                   :: "v"(loff), "v"(g) : "memory");
    }
  }
#else
  {
    float4* dst = reinterpret_cast<float4*>(sP);
    for (int i = 0; i < 7; ++i) {
      int idx = tid + i * 256;
      if (idx < 1664) dst[idx] = src[idx];
    }
  }
#endif

  if (tid < 448) {                 // 28 (padded K) x 16 (N; states in 0..3)
    int k = tid >> 4, n = tid & 15;
    sBT[tid] = (n < 4 && k < An) ? par[PAR_BMAT + n * An + k] : 0.f;
  }
  if (tid == 0 && (int)blockIdx.x + 1 < (int)gridDim.x)
    __builtin_prefetch(x + (posBase + 256) * An, 0, 0);   // global_prefetch_b8

#if defined(__AMDGCN__)
  asm volatile("s_wait_asynccnt 0x0" ::: "memory");  // ASYNCcnt not in barrier
#endif
  __syncthreads();

  float* row = sP + tid * An;
  float m = row[0];
#pragma unroll
  for (int a = 1; a < An; ++a) m = fmaxf(m, row[a]);
  float sum = 0.f;
#pragma unroll
  for (int a = 0; a < An; ++a) { float v = __expf(row[a] - m); sum += v; row[a] = v; }
  sInv[tid] = 1.0f / sum;
  __syncthreads();

#if HMM_USE_WMMA
  const int lane = tid & 31, wv = tid >> 5;
  const int hi = lane >> 4, mrow = lane & 15;
#pragma unroll
  for (int tile = 0; tile < 2; ++tile) {
    const int tb = wv * 32 + tile * 16;      // 16 positions per tile
    v8f acc = {0.f, 0.f, 0.f, 0.f, 0.f, 0.f, 0.f, 0.f};
#pragma unroll
    for (int q = 0; q < 7; ++q) {            // K = 26 padded to 28
      const int k0 = q * 4 + 2 * hi;
      float a0 = (k0     < An) ? sP[(tb + mrow) * An + k0]     : 0.f;
      float a1 = (k0 + 1 < An) ? sP[(tb + mrow) * An + k0 + 1] : 0.f;
      float b0 = sBT[k0 * 16 + mrow];
      float b1 = sBT[(k0 + 1) * 16 + mrow];
      v2f av = {a0, a1}, bv = {b0, b1};
      acc = __builtin_amdgcn_wmma_f32_16x16x4_f32(false, av, false, bv,
                                                  (short)0, acc, false, false);
    }
    if (mrow < 4) {                          // lanes 0-3 / 16-19 own states 0-3
      const int rb = hi ? 8 : 0;
#pragma unroll
      for (int r = 0; r < 8; ++r) {          // acc[r]: rows r (lo) / r+8 (hi)
        const int lp = tb + rb + r;
        const long long gp = posBase + lp;
        le[gp * 4 + mrow] = __logf(acc[r] * sInv[lp] + 1e-16f);
      }
    }
  }
#else
  {
    float d[4] = {0.f, 0.f, 0.f, 0.f};
#pragma unroll
    for (int a = 0; a < An; ++a) {
      float p = row[a];
#pragma unroll
      for (int s = 0; s < 4; ++s) d[s] += p * sBT[a * 16 + s];
    }
    const long long gp = posBase + tid;
#pragma unroll
    for (int s = 0; s < 4; ++s) le[gp * 4 + s] = __logf(d[s] * sInv[tid] + 1e-16f);
  }
#endif
}

// -------------------------------------- phase 1: chunk transfer matrices R_c
// R_c = G_{cL+1} (x) ... (x) G_{(c+1)L}   (LSE semiring; last chunk stops at T-1)
__global__ __launch_bounds__(128) void k_chunkmat(const float* __restrict__ le,
                                                  const float* __restrict__ par,
                                                  float* __restrict__ R) {
  const int tid = blockIdx.x * 128 + threadIdx.x;   // tid = b*Cn + c
  const int b = tid >> 6, c = tid & (Cn - 1);
  float la[16];
#pragma unroll
  for (int i = 0; i < 16; ++i) la[i] = par[PAR_LOGA + i];
  const long long base = (long long)b * Tn;
  const int t0 = c * Ln + 1;
  const int t1 = (c == Cn - 1) ? (Tn - 1) : (c + 1) * Ln;   // inclusive

  float4 e0 = *reinterpret_cast<const float4*>(le + (base + t0) * 4);
  float ev[4] = {e0.x, e0.y, e0.z, e0.w};
  float M[16];
#pragma unroll
  for (int j = 0; j < 4; ++j)
#pragma unroll
    for (int s = 0; s < 4; ++s) M[j * 4 + s] = la[j * 4 + s] + ev[s];

  for (int t = t0 + 1; t <= t1; ++t) {
    float4 et = *reinterpret_cast<const float4*>(le + (base + t) * 4);
    float e[4] = {et.x, et.y, et.z, et.w};
    float Mn[16];
#pragma unroll
    for (int j = 0; j < 4; ++j)
#pragma unroll
      for (int s = 0; s < 4; ++s)
        Mn[j * 4 + s] = lse4(M[j * 4 + 0] + la[0 * 4 + s], M[j * 4 + 1] + la[1 * 4 + s],
                             M[j * 4 + 2] + la[2 * 4 + s], M[j * 4 + 3] + la[3 * 4 + s]) + e[s];
#pragma unroll
    for (int i = 0; i < 16; ++i) M[i] = Mn[i];
  }
  float* out = R + (long long)tid * 16;
#pragma unroll
  for (int i = 0; i < 16; ++i) out[i] = M[i];
}

// -------------------------- phase 2: per-batch boundary scans over 64 chunks
__global__ __launch_bounds__(256) void k_bounds(const float* __restrict__ le,
                                                const float* __restrict__ par,
                                                const float* __restrict__ R,
                                                float* __restrict__ SA,
                                                float* __restrict__ SB) {
  const int b = threadIdx.x;      // one block of 256
  const long long base = (long long)b * Tn;
  // forward: startAlpha[c] = alpha_{cL}
  float4 e0 = *reinterpret_cast<const float4*>(le + base * 4);
  float a[4] = {par[PAR_LOGPI+0] + e0.x, par[PAR_LOGPI+1] + e0.y,
                par[PAR_LOGPI+2] + e0.z, par[PAR_LOGPI+3] + e0.w};
  *reinterpret_cast<float4*>(SA + (long long)(b * Cn) * 4) = make_float4(a[0], a[1], a[2], a[3]);
  for (int c = 0; c < Cn - 1; ++c) {
    const float* Rc = R + (long long)(b * Cn + c) * 16;
    float an[4];
#pragma unroll
    for (int s = 0; s < 4; ++s)
      an[s] = lse4(a[0] + Rc[0 + s], a[1] + Rc[4 + s], a[2] + Rc[8 + s], a[3] + Rc[12 + s]);
#pragma unroll
    for (int s = 0; s < 4; ++s) a[s] = an[s];
    *reinterpret_cast<float4*>(SA + (long long)(b * Cn + c + 1) * 4) =
        make_float4(a[0], a[1], a[2], a[3]);
  }
  // backward: startBeta[c] = beta_{cL};  beta_{T-1} = 0
  const float* Rl = R + (long long)(b * Cn + Cn - 1) * 16;
  float sb[4];
#pragma unroll
  for (int j = 0; j < 4; ++j) sb[j] = lse4(Rl[j*4+0], Rl[j*4+1], Rl[j*4+2], Rl[j*4+3]);
  *reinterpret_cast<float4*>(SB + (long long)(b * Cn + Cn - 1) * 4) =
      make_float4(sb[0], sb[1], sb[2], sb[3]);
  for (int c = Cn - 2; c >= 0; --c) {
    const float* Rc = R + (long long)(b * Cn + c) * 16;
    float sn[4];
#pragma unroll
    for (int j = 0; j < 4; ++j)
      sn[j] = lse4(Rc[j*4+0] + sb[0], Rc[j*4+1] + sb[1], Rc[j*4+2] + sb[2], Rc[j*4+3] + sb[3]);
#pragma unroll
    for (int j = 0; j < 4; ++j) sb[j] = sn[j];
    *reinterpret_cast<float4*>(SB + (long long)(b * Cn + c) * 4) =
        make_float4(sb[0], sb[1], sb[2], sb[3]);
  }
}

// ------------------------------------ phase 3a: replay alphas inside chunks
__global__ __launch_bounds__(128) void k_alphas(const float* __restrict__ le,
                                                const float* __restrict__ par,
                                                const float* __restrict__ SA,
                                                float* __restrict__ AL) {
  const int tid = blockIdx.x * 128 + threadIdx.x;
  const int b = tid >> 6, c = tid & (Cn - 1);
  float la[16];
#pragma unroll
  for (int i = 0; i < 16; ++i) la[i] = par[PAR_LOGA + i];
  const long long base = (long long)b * Tn;
  const int t0 = c * Ln;
  float4 av = *reinterpret_cast<const float4*>(SA + (long long)(b * Cn + c) * 4);
  float a[4] = {av.x, av.y, av.z, av.w};
  *reinterpret_cast<float4*>(AL + (base + t0) * 4) = av;
  for (int t = t0 + 1; t < t0 + Ln; ++t) {
    float4 et = *reinterpret_cast<const float4*>(le + (base + t) * 4);
    float e[4] = {et.x, et.y, et.z, et.w};
    float an[4];
#pragma unroll
    for (int s = 0; s < 4; ++s)
      an[s] = e[s] + lse4(a[0] + la[0 + s], a[1] + la[4 + s], a[2] + la[8 + s], a[3] + la[12 + s]);
#pragma unroll
    for (int s = 0; s < 4; ++s) a[s] = an[s];
    *reinterpret_cast<float4*>(AL + (base + t) * 4) = make_float4(a[0], a[1], a[2], a[3]);
  }
}

// ------------ phase 3b: betas backward + gamma softmax + 4x5 projection + out
__global__ __launch_bounds__(128) void k_output(const float* __restrict__ le,
                                                const float* __restrict__ par,
                                                const float* __restrict__ AL,
                                                const float* __restrict__ SB,
                                                float* __restrict__ out) {
  const int tid = blockIdx.x * 128 + threadIdx.x;
  const int b = tid >> 6, c = tid & (Cn - 1);
  float la[16], W[20];
#pragma unroll
  for (int i = 0; i < 16; ++i) la[i] = par[PAR_LOGA + i];
#pragma unroll
  for (int i = 0; i < 20; ++i) W[i] = par[PAR_WOUT + i];
  const long long base = (long long)b * Tn;
  const int t0 = c * Ln, tEnd = t0 + Ln - 1;

  float bt[4];
  if (c == Cn - 1) {
    bt[0] = bt[1] = bt[2] = bt[3] = 0.f;
  } else {                                   // beta_{(c+1)L-1} = G_{(c+1)L} . startBeta[c+1]
    float4 et = *reinterpret_cast<const float4*>(le + (base + (c + 1) * Ln) * 4);
    float4 sv = *reinterpret_cast<const float4*>(SB + (long long)(b * Cn + c + 1) * 4);
    float e[4] = {et.x, et.y, et.z, et.w};
    float s[4] = {sv.x, sv.y, sv.z, sv.w};
#pragma unroll
    for (int j = 0; j < 4; ++j)
      bt[j] = lse4(la[j*4+0] + e[0] + s[0], la[j*4+1] + e[1] + s[1],
                   la[j*4+2] + e[2] + s[2], la[j*4+3] + e[3] + s[3]);
  }

  for (int t = tEnd; t >= t0; --t) {
    float4 av = *reinterpret_cast<const float4*>(AL + (base + t) * 4);
    float v[4] = {av.x + bt[0], av.y + bt[1], av.z + bt[2], av.w + bt[3]};
    float m = fmaxf(fmaxf(v[0], v[1]), fmaxf(v[2], v[3]));
    float g0 = __expf(v[0]-m), g1 = __expf(v[1]-m), g2 = __expf(v[2]-m), g3 = __expf(v[3]-m);
    float inv = 1.0f / (g0 + g1 + g2 + g3);
    float* o = out + (base + t) * 5;
#pragma unroll
    for (int k = 0; k < 5; ++k)
      o[k] = inv * (g0 * W[0 * 5 + k] + g1 * W[1 * 5 + k] + g2 * W[2 * 5 + k] + g3 * W[3 * 5 + k]);
    if (t > t0) {                            // beta_{t-1} = G_t . beta_t
      float4 et = *reinterpret_cast<const float4*>(le + (base + t) * 4);
      float e[4] = {et.x, et.y, et.z, et.w};
      float bn[4];
#pragma unroll
      for (int j = 0; j < 4; ++j)
        bn[j] = lse4(la[j*4+0] + e[0] + bt[0], la[j*4+1] + e[1] + bt[1],
                     la[j*4+2] + e[2] + bt[2], la[j*4+3] + e[3] + bt[3]);
#pragma unroll
      for (int j = 0; j < 4; ++j) bt[j] = bn[j];
    }
  }
}

extern "C" void kernel_launch(void* const* d_in, const int* in_sizes, int n_in,
                              void* d_out, int out_size, void* d_ws, size_t ws_size,
                              hipStream_t stream) {
  (void)in_sizes; (void)n_in; (void)out_size; (void)ws_size;
  const float* x     = (const float*)d_in[0];
  const float* emit  = (const float*)d_in[1];
  const float* trans = (const float*)d_in[2];
  const float* init  = (const float*)d_in[3];
  const float* wout  = (const float*)d_in[4];
  float* ws  = (float*)d_ws;
  float* out = (float*)d_out;

  k_setup<<<1, 64, 0, stream>>>(emit, trans, init, wout, ws);
  k_emission<<<(Bn * Tn) / 256, 256, 0, stream>>>(x, ws, ws + OFF_LE);
  k_chunkmat<<<(Bn * Cn) / 128, 128, 0, stream>>>(ws + OFF_LE, ws, ws + OFF_R);
  k_bounds<<<1, 256, 0, stream>>>(ws + OFF_LE, ws, ws + OFF_R, ws + OFF_SA, ws + OFF_SB);
  k_alphas<<<(Bn * Cn) / 128, 128, 0, stream>>>(ws + OFF_LE, ws, ws + OFF_SA, ws + OFF_AL);
  k_output<<<(Bn * Cn) / 128, 128, 0, stream>>>(ws + OFF_LE, ws, ws + OFF_AL, ws + OFF_SB, out);
}